// PhasedGNN_48610439856172
// MI455X (gfx1250) — compile-verified
//
#include <hip/hip_runtime.h>
#include <hip/hip_bf16.h>

// ---------------- problem constants ----------------
#define NN      100000      // nodes
#define NE      3200000     // edges
#define NGRAPH  64
#define INDIM   128
#define HID     64
#define NFAM    64
#define CPF     3
#define NGOOD   5
#define BN_EPS  1e-5f

typedef float v2f __attribute__((ext_vector_type(2)));
typedef float v8f __attribute__((ext_vector_type(8)));

// ---------------- small helpers ----------------
__global__ void k_fill(float* p, int n, float v) {
    int i = blockIdx.x * blockDim.x + threadIdx.x;
    if (i < n) p[i] = v;
}

__global__ void k_deg_accum(const int* __restrict__ cols, float* __restrict__ deg) {
    int e = blockIdx.x * blockDim.x + threadIdx.x;
    if (e < NE) atomicAdd(&deg[cols[e]], 1.0f);
}

__global__ void k_rsqrt_inplace(float* p, int n) {
    int i = blockIdx.x * blockDim.x + threadIdx.x;
    if (i < n) p[i] = rsqrtf(p[i]);
}

// ---------------- dense GEMM via f32 WMMA ----------------
// out[NN,64] = A[NN,K] @ W[K,64]  using V_WMMA_F32_16X16X4_F32.
// One wave computes one 16x16 output tile. NN is a multiple of 16 so every
// tile is full and EXEC stays all-ones across the WMMA (ISA requirement).
//
// Fragment layouts (ISA 7.12.2, 32-bit):
//   A 16x4 : lanes 0-15 -> M=lane, VGPR0=K0 VGPR1=K1; lanes 16-31 -> K2/K3
//   B 4x16 : lanes hold N, VGPR holds K, lane-half selects K pair (mirror of A)
//   C/D    : VGPR j: lanes 0-15 -> M=j, N=lane; lanes 16-31 -> M=j+8, N=lane-16
__global__ void k_gemm_wmma(const float* __restrict__ A,
                            const float* __restrict__ W,
                            float* __restrict__ out, int K) {
    const int wavesPerBlock = blockDim.x >> 5;
    int wave = blockIdx.x * wavesPerBlock + (threadIdx.x >> 5);
    int lane = threadIdx.x & 31;
    int tileM = wave >> 2;          // HID/16 = 4 column tiles
    int tileN = wave & 3;
    if (tileM * 16 >= NN) return;   // uniform per wave

    int m      = tileM * 16 + (lane & 15);
    int n      = tileN * 16 + (lane & 15);
    int khalf  = (lane >> 4) << 1;  // 0 for lanes 0-15, 2 for lanes 16-31

    v8f acc = {};
    for (int k = 0; k < K; k += 4) {
        v2f a, b;
        a.x = A[m * K + k + khalf];
        a.y = A[m * K + k + khalf + 1];
        b.x = W[(k + khalf)     * HID + n];
        b.y = W[(k + khalf + 1) * HID + n];
        acc = __builtin_amdgcn_wmma_f32_16x16x4_f32(
                  false, a, false, b, (short)0, acc, false, false);
    }

    int mout = tileM * 16 + ((lane >> 4) << 3);
    int nout = tileN * 16 + (lane & 15);
#pragma unroll
    for (int j = 0; j < 8; ++j)
        out[(mout + j) * HID + nout] = acc[j];
}

// ---------------- graph aggregation ----------------
// agg[n][f] = h[n][f] * dinv[n]^2     (self-loop term; also zero-inits agg)
__global__ void k_init_agg(const float* __restrict__ h,
                           const float* __restrict__ dinv,
                           float* __restrict__ agg) {
    int idx = blockIdx.x * blockDim.x + threadIdx.x;   // exactly NN*HID threads
    int n = idx >> 6;
    float d = dinv[n];
    agg[idx] = h[idx] * d * d;
}

// agg[col] += h[row] * dinv[row]*dinv[col]   (one wave per edge strip;
// lanes cover 64 features as float2 -> coalesced 256B gather per edge)
__global__ void k_scatter_edges(const int* __restrict__ rows,
                                const int* __restrict__ cols,
                                const float* __restrict__ dinv,
                                const float* __restrict__ h,
                                float* __restrict__ agg) {
    int lane  = threadIdx.x & 31;
    int wave  = blockIdx.x * (blockDim.x >> 5) + (threadIdx.x >> 5);
    int nwave = gridDim.x * (blockDim.x >> 5);
    const float2* h2 = (const float2*)h;
    for (int e = wave; e < NE; e += nwave) {
        int r = rows[e], c = cols[e];
        float en = dinv[r] * dinv[c];
        float2 v = h2[r * 32 + lane];
        atomicAdd(&agg[c * HID + 2 * lane],     v.x * en);
        atomicAdd(&agg[c * HID + 2 * lane + 1], v.y * en);
    }
}

// ---------------- batch-norm ----------------
// stats[f] = sum, stats[64+f] = sum of squares (over NN rows)
__global__ void k_bn_stats(const float* __restrict__ agg, float* __restrict__ stats) {
    int f = threadIdx.x & 63;
    int r = threadIdx.x >> 6;                 // 4 row-groups per block
    float s = 0.f, sq = 0.f;
    for (int n = blockIdx.x * 4 + r; n < NN; n += gridDim.x * 4) {
        float v = agg[n * HID + f];
        s += v; sq += v * v;
    }
    atomicAdd(&stats[f], s);
    atomicAdd(&stats[64 + f], sq);
}

// out = (x-m)*gamma*rsqrt(var+eps)+beta, optional ReLU. (bias b is a no-op
// under training-mode BN: the mean absorbs it, so it is omitted.)
__global__ void k_bn_apply(const float* __restrict__ agg,
                           const float* __restrict__ stats,
                           const float* __restrict__ gamma,
                           const float* __restrict__ beta,
                           float* __restrict__ out, int relu) {
    int idx = blockIdx.x * blockDim.x + threadIdx.x;   // NN*HID threads
    int f = idx & 63;
    const float invn = 1.0f / (float)NN;
    float m   = stats[f] * invn;
    float var = stats[64 + f] * invn - m * m;
    float sc  = gamma[f] * rsqrtf(var + BN_EPS);
    float v   = (agg[idx] - m) * sc + beta[f];
    out[idx]  = relu ? fmaxf(v, 0.0f) : v;
}

// ---------------- pooling ----------------
__global__ void k_pool_accum(const float* __restrict__ hin,
                             const int* __restrict__ batch,
                             float* __restrict__ psum) {
    int idx = blockIdx.x * blockDim.x + threadIdx.x;   // NN*HID threads
    int n = idx >> 6, f = idx & 63;
    atomicAdd(&psum[batch[n] * HID + f], hin[idx]);
}

__global__ void k_pool_cnt(const int* __restrict__ batch, float* __restrict__ pcnt) {
    int n = blockIdx.x * blockDim.x + threadIdx.x;
    if (n < NN) atomicAdd(&pcnt[batch[n]], 1.0f);
}

__global__ void k_pool_fin(const float* __restrict__ psum,
                           const float* __restrict__ pcnt,
                           float* __restrict__ emb) {
    int idx = blockIdx.x * blockDim.x + threadIdx.x;   // NGRAPH*HID
    if (idx < NGRAPH * HID) {
        int g = idx >> 6;
        emb[idx] = psum[idx] / fmaxf(pcnt[g], 1.0f);
    }
}

// ---------------- centroid classifier ----------------
__global__ void k_centroid(const float* __restrict__ emb,
                           const float* __restrict__ cg,
                           const float* __restrict__ cm,
                           const float* __restrict__ temp,
                           float* __restrict__ out) {
    int g = threadIdx.x;
    if (g >= NGRAPH) return;
    const float* e = emb + g * HID;
    float it = 1.0f / temp[0];

    float ming = 3.4e38f;
    for (int c = 0; c < NGOOD; ++c) {
        const float* cr = cg + c * HID;
        float d = 0.f;
        for (int k = 0; k < HID; ++k) { float t = e[k] - cr[k]; d += t * t; }
        ming = fminf(ming, d);
    }
    out[g * (1 + NFAM)] = -ming * it;

    for (int fam = 0; fam < NFAM; ++fam) {
        float mm = 3.4e38f;
        for (int c = 0; c < CPF; ++c) {
            const float* cr = cm + (fam * CPF + c) * HID;
            float d = 0.f;
            for (int k = 0; k < HID; ++k) { float t = e[k] - cr[k]; d += t * t; }
            mm = fminf(mm, d);
        }
        out[g * (1 + NFAM) + 1 + fam] = -mm * it;
    }
}

// ---------------- launcher ----------------
extern "C" void kernel_launch(void* const* d_in, const int* in_sizes, int n_in,
                              void* d_out, int out_size, void* d_ws, size_t ws_size,
                              hipStream_t stream) {
    const float* x     = (const float*)d_in[0];
    const int*   ei    = (const int*)  d_in[1];   // [2, NE]: rows then cols
    const int*   batch = (const int*)  d_in[2];
    const float* W1 = (const float*)d_in[3];
    const float* g1 = (const float*)d_in[5];
    const float* be1= (const float*)d_in[6];
    const float* W2 = (const float*)d_in[7];
    const float* g2 = (const float*)d_in[9];
    const float* be2= (const float*)d_in[10];
    const float* W3 = (const float*)d_in[11];
    const float* g3 = (const float*)d_in[13];
    const float* be3= (const float*)d_in[14];
    const float* cg = (const float*)d_in[15];
    const float* cm = (const float*)d_in[16];
    const float* temp = (const float*)d_in[17];
    float* out = (float*)d_out;

    // workspace layout (floats)
    float* ws    = (float*)d_ws;
    float* deg   = ws;                        // NN (becomes dinv_sqrt in place)
    float* h     = deg  + 100096;             // NN*HID gemm output
    float* agg   = h    + NN * HID;           // NN*HID aggregation
    float* hin   = agg  + NN * HID;           // NN*HID normalized activations
    float* stats = hin  + NN * HID;           // 128 (sum, sumsq)
    float* psum  = stats + 128;               // NGRAPH*HID
    float* pcnt  = psum + NGRAPH * HID;       // NGRAPH
    float* emb   = pcnt + NGRAPH;             // NGRAPH*HID

    const int TB = 256;
    const int nodeBlk  = (NN + TB - 1) / TB;          // 391
    const int elemBlk  = (NN * HID) / TB;             // 25000 (exact)
    const int edgeBlk  = (NE + TB - 1) / TB;          // 12500 (exact)
    const int gemmBlk  = ((NN / 16) * 4) / 8;         // 3125 (8 waves/block)

    // degrees (with self loop) -> dinv_sqrt
    k_fill<<<nodeBlk, TB, 0, stream>>>(deg, NN, 1.0f);
    k_deg_accum<<<edgeBlk, TB, 0, stream>>>(ei + NE, deg);
    k_rsqrt_inplace<<<nodeBlk, TB, 0, stream>>>(deg, NN);

    auto layer = [&](const float* in, int K, const float* W,
                     const float* gma, const float* bta, int relu) {
        k_gemm_wmma<<<gemmBlk, TB, 0, stream>>>(in, W, h, K);
        k_init_agg<<<elemBlk, TB, 0, stream>>>(h, deg, agg);
        k_scatter_edges<<<4096, TB, 0, stream>>>(ei, ei + NE, deg, h, agg);
        k_fill<<<1, 128, 0, stream>>>(stats, 128, 0.0f);
        k_bn_stats<<<1024, TB, 0, stream>>>(agg, stats);
        k_bn_apply<<<elemBlk, TB, 0, stream>>>(agg, stats, gma, bta, hin, relu);
    };
    layer(x,   INDIM, W1, g1, be1, 1);
    layer(hin, HID,   W2, g2, be2, 1);
    layer(hin, HID,   W3, g3, be3, 0);

    // global mean pool
    k_fill<<<(NGRAPH * HID + NGRAPH + TB - 1) / TB, TB, 0, stream>>>(
        psum, NGRAPH * HID + NGRAPH, 0.0f);          // psum + pcnt contiguous
    k_pool_accum<<<elemBlk, TB, 0, stream>>>(hin, batch, psum);
    k_pool_cnt<<<nodeBlk, TB, 0, stream>>>(batch, pcnt);
    k_pool_fin<<<(NGRAPH * HID + TB - 1) / TB, TB, 0, stream>>>(psum, pcnt, emb);

    // centroid logits [64, 65]
    k_centroid<<<1, 64, 0, stream>>>(emb, cg, cm, temp, out);
}